// GraphEncoder_87497073754991
// MI455X (gfx1250) — compile-verified
//
#include <hip/hip_runtime.h>

#define N_NODES 65536
#define N_EDGES 1048576
#define DIM 64

typedef float v2f __attribute__((ext_vector_type(2)));
typedef float v8f __attribute__((ext_vector_type(8)));

// ---------------------------------------------------------------- utilities

__global__ __launch_bounds__(256) void zero_u32_kernel(unsigned* __restrict__ p, int n) {
    int i = blockIdx.x * 256 + threadIdx.x;
    if (i < n) p[i] = 0u;
}

// one thread per edge: histogram out/in degrees (u32 atomics, done once)
__global__ __launch_bounds__(256) void degree_kernel(const int* __restrict__ ei,
                                                     unsigned* __restrict__ deg_out,
                                                     unsigned* __restrict__ deg_in) {
    int e = blockIdx.x * 256 + threadIdx.x;
    if (e < N_EDGES) {
        atomicAdd(&deg_out[ei[e]], 1u);
        atomicAdd(&deg_in[ei[N_EDGES + e]], 1u);
    }
}

__global__ __launch_bounds__(256) void rsqrt_kernel(const unsigned* __restrict__ deg_out,
                                                    const unsigned* __restrict__ deg_in,
                                                    float* __restrict__ rs_out,
                                                    float* __restrict__ rs_in) {
    int i = blockIdx.x * 256 + threadIdx.x;
    if (i < N_NODES) {
        unsigned dovr = deg_out[i]; if (dovr < 1u) dovr = 1u;
        unsigned din  = deg_in[i];  if (din  < 1u) din  = 1u;
        rs_out[i] = rsqrtf((float)dovr);
        rs_in[i]  = rsqrtf((float)din);
    }
}

// single-workgroup exclusive scan of in-degrees -> CSR row_ptr (65537 entries)
__global__ __launch_bounds__(1024) void scan_kernel(const unsigned* __restrict__ deg_in,
                                                    int* __restrict__ row_ptr) {
    __shared__ unsigned part[1024];
    int t = threadIdx.x;
    int base = t * (N_NODES / 1024);            // 64 nodes per thread
    unsigned sum = 0;
    for (int i = 0; i < N_NODES / 1024; ++i) sum += deg_in[base + i];
    part[t] = sum;
    __syncthreads();
    // Hillis-Steele inclusive scan over 1024 partials
    for (int off = 1; off < 1024; off <<= 1) {
        unsigned v = (t >= off) ? part[t - off] : 0u;
        __syncthreads();
        part[t] += v;
        __syncthreads();
    }
    unsigned run = (t == 0) ? 0u : part[t - 1];  // exclusive prefix of my chunk
    for (int i = 0; i < N_NODES / 1024; ++i) {
        row_ptr[base + i] = (int)run;
        run += deg_in[base + i];
    }
    if (t == 1023) row_ptr[N_NODES] = (int)run;  // == N_EDGES
}

// one thread per edge: bin sources into per-dst CSR slots
__global__ __launch_bounds__(256) void fill_kernel(const int* __restrict__ ei,
                                                   unsigned* __restrict__ cursor,
                                                   const int* __restrict__ row_ptr,
                                                   int* __restrict__ in_src) {
    int e = blockIdx.x * 256 + threadIdx.x;
    if (e < N_EDGES) {
        int s = ei[e];
        int d = ei[N_EDGES + e];
        unsigned slot = atomicAdd(&cursor[d], 1u);
        in_src[row_ptr[d] + (int)slot] = s;
    }
}

// hs = x * rs_out  (float4 vectorized, 16 float4 per row)
__global__ __launch_bounds__(256) void scale0_kernel(const float* __restrict__ x,
                                                     const float* __restrict__ rs_out,
                                                     float* __restrict__ hs) {
    int t = blockIdx.x * 256 + threadIdx.x;           // t < N_NODES*DIM/4
    float4 v = ((const float4*)x)[t];
    float r = rs_out[t >> 4];
    v.x *= r; v.y *= r; v.z *= r; v.w *= r;
    ((float4*)hs)[t] = v;
}

// One wave per destination node. 32 lanes each own a float2 of the 64-wide
// feature row. Incoming-edge indices read 32-at-a-time, broadcast via shuffle,
// then every lane does a fully coalesced 256B-row float2 gather. No atomics.
// Output is pre-scaled by rsqrt(in-degree).
__global__ __launch_bounds__(256) void agg_kernel(const float* __restrict__ hs,
                                                  const int* __restrict__ in_src,
                                                  const int* __restrict__ row_ptr,
                                                  const float* __restrict__ rs_in,
                                                  float* __restrict__ aggn) {
    int node = (blockIdx.x * 256 + threadIdx.x) >> 5;
    int lane = threadIdx.x & 31;
    int start = row_ptr[node];
    int end   = row_ptr[node + 1];
    float ax = 0.0f, ay = 0.0f;
    for (int base = start; base < end; base += 32) {
        int i = base + lane;
        int idx = (i < end) ? in_src[i] : 0;
        int cnt = end - base; if (cnt > 32) cnt = 32;
        for (int j = 0; j < cnt; ++j) {
            int s = __shfl(idx, j, 32);
            const v2f v = *(const v2f*)(hs + (size_t)s * DIM + lane * 2);
            ax += v.x; ay += v.y;
        }
    }
    float r = rs_in[node];
    v2f outv; outv.x = ax * r; outv.y = ay * r;
    *(v2f*)(aggn + (size_t)node * DIM + lane * 2) = outv;
}

// Y[65536x64] = act(X @ W + b) [* rs_out per-row if relu_scale]
// 4 waves / block, each wave: 16 rows x 64 cols via V_WMMA_F32_16X16X4_F32.
// A frag (16x4 fp32): lanes 0-15 hold K={k,k+1}, lanes 16-31 hold K={k+2,k+3}.
// W is pre-swizzled into LDS in exact B-fragment order, so each B fragment is
// one conflict-free ds_load_b64 into an even-aligned VGPR pair (no v_mov fixup).
__global__ __launch_bounds__(128) void gemm_wmma_kernel(const float* __restrict__ X,
                                                        const float* __restrict__ W,
                                                        const float* __restrict__ bias,
                                                        const float* __restrict__ rs_out,
                                                        float* __restrict__ Y,
                                                        int relu_scale) {
    // lWsw[((f*4 + t)*32 + lane)*2 + {0,1}] = { W[ka][n], W[ka+1][n] }
    //   f = k/4 (16 steps), t = N-tile (4), ka = 4f + 2*(lane>>4), n = 16t + (lane&15)
    __shared__ float lWsw[16 * 4 * 32 * 2];   // 16 KB
    __shared__ float lB[DIM];
    for (int p = threadIdx.x; p < 16 * 4 * 32; p += 128) {
        int l  = p & 31;
        int t  = (p >> 5) & 3;
        int f  = p >> 7;
        int ka = 4 * f + 2 * (l >> 4);
        int n  = 16 * t + (l & 15);
        v2f w2;
        w2.x = W[ka * DIM + n];
        w2.y = W[(ka + 1) * DIM + n];
        *(v2f*)&lWsw[p * 2] = w2;
    }
    if (threadIdx.x < DIM) lB[threadIdx.x] = bias[threadIdx.x];
    __syncthreads();

    const int wave = threadIdx.x >> 5;
    const int lane = threadIdx.x & 31;
    const int half = lane >> 4;          // 0: lanes 0-15, 1: lanes 16-31
    const int l16  = lane & 15;
    const int m0   = blockIdx.x * 64 + wave * 16;
    const int mA   = m0 + l16;           // A-matrix row owned by this lane

    v8f acc[4] = {v8f{}, v8f{}, v8f{}, v8f{}};

#pragma unroll
    for (int f = 0; f < 16; ++f) {
        const int ka = 4 * f + 2 * half;
        v2f a = *(const v2f*)(X + (size_t)mA * DIM + ka);
#pragma unroll
        for (int t = 0; t < 4; ++t) {
            v2f b = *(const v2f*)&lWsw[(((f * 4 + t) * 32) + lane) * 2];
            acc[t] = __builtin_amdgcn_wmma_f32_16x16x4_f32(
                false, a, false, b, (short)0, acc[t], false, false);
        }
    }

    // C/D layout: VGPR r -> row m0 + half*8 + r, col = tile*16 + l16
    const int mrow0 = m0 + half * 8;
    float rsv[8];
#pragma unroll
    for (int r = 0; r < 8; ++r) rsv[r] = relu_scale ? rs_out[mrow0 + r] : 1.0f;

#pragma unroll
    for (int t = 0; t < 4; ++t) {
        const int n = t * 16 + l16;
        const float bn = lB[n];
#pragma unroll
        for (int r = 0; r < 8; ++r) {
            float v = acc[t][r] + bn;
            if (relu_scale) v = fmaxf(v, 0.0f) * rsv[r];
            Y[(size_t)(mrow0 + r) * DIM + n] = v;
        }
    }
}

extern "C" void kernel_launch(void* const* d_in, const int* in_sizes, int n_in,
                              void* d_out, int out_size, void* d_ws, size_t ws_size,
                              hipStream_t stream) {
    (void)in_sizes; (void)n_in; (void)out_size; (void)ws_size;
    const float* x  = (const float*)d_in[0];
    const int*   ei = (const int*)d_in[1];
    // d_in[2] = batch_num_nodes: unbatch is the identity for this shape
    const float* W1 = (const float*)d_in[3];
    const float* b1 = (const float*)d_in[4];
    const float* W2 = (const float*)d_in[5];
    const float* b2 = (const float*)d_in[6];
    const float* W3 = (const float*)d_in[7];
    const float* b3 = (const float*)d_in[8];
    float* out = (float*)d_out;

    char* ws = (char*)d_ws;
    unsigned* deg_out = (unsigned*)ws; ws += (size_t)N_NODES * 4;
    unsigned* deg_in  = (unsigned*)ws; ws += (size_t)N_NODES * 4;
    unsigned* cursor  = (unsigned*)ws; ws += (size_t)N_NODES * 4;   // contiguous with deg_*
    float*    rs_out  = (float*)ws;    ws += (size_t)N_NODES * 4;
    float*    rs_in   = (float*)ws;    ws += (size_t)N_NODES * 4;
    int*      row_ptr = (int*)ws;      ws += (size_t)(N_NODES + 16) * 4;
    int*      in_src  = (int*)ws;      ws += (size_t)N_EDGES * 4;
    float*    hs      = (float*)ws;    ws += (size_t)N_NODES * DIM * 4;
    float*    aggn    = (float*)ws;    ws += (size_t)N_NODES * DIM * 4;

    // ---- graph preprocessing (once per call; graph is static)
    zero_u32_kernel<<<(3 * N_NODES + 255) / 256, 256, 0, stream>>>(deg_out, 3 * N_NODES);
    degree_kernel<<<N_EDGES / 256, 256, 0, stream>>>(ei, deg_out, deg_in);
    rsqrt_kernel<<<N_NODES / 256, 256, 0, stream>>>(deg_out, deg_in, rs_out, rs_in);
    scan_kernel<<<1, 1024, 0, stream>>>(deg_in, row_ptr);
    fill_kernel<<<N_EDGES / 256, 256, 0, stream>>>(ei, cursor, row_ptr, in_src);

    // ---- layer 0 input scaling: hs = x * rsqrt(out-degree)
    scale0_kernel<<<(N_NODES * DIM / 4) / 256, 256, 0, stream>>>(x, rs_out, hs);

    // ---- layer 1
    agg_kernel<<<(N_NODES * 32) / 256, 256, 0, stream>>>(hs, in_src, row_ptr, rs_in, aggn);
    gemm_wmma_kernel<<<N_NODES / 64, 128, 0, stream>>>(aggn, W1, b1, rs_out, hs, 1);
    // ---- layer 2
    agg_kernel<<<(N_NODES * 32) / 256, 256, 0, stream>>>(hs, in_src, row_ptr, rs_in, aggn);
    gemm_wmma_kernel<<<N_NODES / 64, 128, 0, stream>>>(aggn, W2, b2, rs_out, hs, 1);
    // ---- layer 3 (no ReLU, no rescale) -> final output (unbatch == identity)
    agg_kernel<<<(N_NODES * 32) / 256, 256, 0, stream>>>(hs, in_src, row_ptr, rs_in, aggn);
    gemm_wmma_kernel<<<N_NODES / 64, 128, 0, stream>>>(aggn, W3, b3, rs_out, out, 0);
}